// RankNetFast_2491081031806
// MI455X (gfx1250) — compile-verified
//
#include <hip/hip_runtime.h>

// RankNetFast reduced form:
//   row_i = -0.5*SIGMA*(less(t_i) - greater(t_i))   (from 5-bin histogram of targets)
//   loss  = sum_{j<=k} row_j * s_k                  (triangular bilinear form, n=8192)
// Blocked 16-wide:
//   loss = sum_I D_I + sum_{I<J} R_I * S_J
//   D_I  = r_I^T U s_I  computed via  T = Rmat x U  with V_WMMA_F32_16X16X4_F32 (x4, K-chunked)
// Single workgroup (1024 thr = 32 waves), ~45KB static LDS. Launch-latency bound by design.

#define N_SAMP 8192
#define SIGMA  1.0f

typedef __attribute__((ext_vector_type(2))) float v2f;
typedef __attribute__((ext_vector_type(8))) float v8f;

// Map a target value 0..4 to its row value without dynamic register indexing.
__device__ __forceinline__ float rowOf(int v, float r0, float r1, float r2, float r3, float r4) {
    float x = r0;
    x = (v == 1) ? r1 : x;
    x = (v == 2) ? r2 : x;
    x = (v == 3) ? r3 : x;
    x = (v == 4) ? r4 : x;
    return x;
}

__global__ __launch_bounds__(1024, 1)
void ranknet_fused_kernel(const float* __restrict__ scores,
                          const int*   __restrict__ target,
                          const int*   __restrict__ docs,
                          float*       __restrict__ out) {
    __shared__ float         s_lds[N_SAMP];   // 32 KB: gathered scores
    __shared__ unsigned char t8[N_SAMP];      //  8 KB: gathered targets (0..4)
    __shared__ float         Rb[512];         //  2 KB: per-16-block row sums
    __shared__ float         Sb[512];         //  2 KB: per-16-block score sums
    __shared__ int           hist[8];
    __shared__ float         wpart[32];       // per-wave diagonal partials

    const int tid  = threadIdx.x;
    const int lane = tid & 31;
    const int wave = tid >> 5;

    if (tid < 8) hist[tid] = 0;
    __syncthreads();

    // ---- Phase A: gather (coalesced docs read, random gathers) + histogram ----
    int c0 = 0, c1 = 0, c2 = 0, c3 = 0, c4 = 0;
#pragma unroll
    for (int t = 0; t < 8; ++t) {
        const int j  = tid + 1024 * t;
        const int d  = docs[j];
        const int tv = target[d];
        const float sv = scores[d];         // scores is [N_DOCS,1]
        s_lds[j] = sv;
        t8[j]    = (unsigned char)tv;
        c0 += (tv == 0); c1 += (tv == 1); c2 += (tv == 2); c3 += (tv == 3); c4 += (tv == 4);
    }
    atomicAdd(&hist[0], c0);
    atomicAdd(&hist[1], c1);
    atomicAdd(&hist[2], c2);
    atomicAdd(&hist[3], c3);
    atomicAdd(&hist[4], c4);
    __syncthreads();

    // ---- Phase B: per-class row values from the histogram ----
    const int h0 = hist[0], h1 = hist[1], h2 = hist[2], h3 = hist[3], h4 = hist[4];
    const int l0 = 0, l1 = h0, l2 = h0 + h1, l3 = l2 + h2, l4 = l3 + h3;
    // greater_v = N - less_v - h_v ;  row(v) = -0.5*SIGMA*(less - greater)
    const float r0 = -0.5f * SIGMA * (float)(l0 - (N_SAMP - l0 - h0));
    const float r1 = -0.5f * SIGMA * (float)(l1 - (N_SAMP - l1 - h1));
    const float r2 = -0.5f * SIGMA * (float)(l2 - (N_SAMP - l2 - h2));
    const float r3 = -0.5f * SIGMA * (float)(l3 - (N_SAMP - l3 - h3));
    const float r4 = -0.5f * SIGMA * (float)(l4 - (N_SAMP - l4 - h4));

    // ---- Phase C: per-wave diagonal blocks via WMMA f32 16x16x4 (x4 K-chunks) ----
    // Wave w owns elements [256w, 256w+256): Rmat[I][a] = row(t8[256w+16I+a]),
    // U[a][b] = (a<=b).  T = Rmat*U ;  diag contribution = sum T .* Smat.
    {
        const int base = wave << 8;               // 256*wave
        const int idx  = lane & 15;               // A row m == B col n
        const int kb   = (lane < 16) ? 0 : 2;     // K half-split (A 16x4 and B 4x16 layouts)
        v8f acc = {0.f, 0.f, 0.f, 0.f, 0.f, 0.f, 0.f, 0.f};
#pragma unroll
        for (int c = 0; c < 4; ++c) {
            const int k0 = 4 * c + kb;
            const int k1 = k0 + 1;
            v2f a, b;
            a.x = rowOf((int)t8[base + 16 * idx + k0], r0, r1, r2, r3, r4);
            a.y = rowOf((int)t8[base + 16 * idx + k1], r0, r1, r2, r3, r4);
            b.x = (k0 <= idx) ? 1.0f : 0.0f;      // U[k0][n]
            b.y = (k1 <= idx) ? 1.0f : 0.0f;      // U[k1][n]
            acc = __builtin_amdgcn_wmma_f32_16x16x4_f32(
                /*neg_a=*/false, a, /*neg_b=*/false, b,
                /*c_mod=*/(short)0, acc, /*reuse_a=*/false, /*reuse_b=*/false);
        }
        // D layout: lane n (0-15) vgpr v -> T[v][n]; lane n+16 vgpr v -> T[v+8][n]
        const int rb = (lane < 16) ? 0 : 8;
        float part = 0.f;
#pragma unroll
        for (int v = 0; v < 8; ++v)
            part += acc[v] * s_lds[base + 16 * (v + rb) + idx];
#pragma unroll
        for (int off = 16; off >= 1; off >>= 1)
            part += __shfl_xor(part, off, 32);
        if (lane == 0) wpart[wave] = part;
    }

    // ---- Phase D: 16-block sums R_I, S_I ----
    if (tid < 512) {
        const int I = tid;
        float rs = 0.f;
#pragma unroll
        for (int a = 0; a < 16; ++a)
            rs += rowOf((int)t8[(I << 4) + a], r0, r1, r2, r3, r4);
        Rb[I] = rs;
    } else {
        const int I = tid - 512;
        float ss = 0.f;
#pragma unroll
        for (int b = 0; b < 16; ++b)
            ss += s_lds[(I << 4) + b];
        Sb[I] = ss;
    }
    __syncthreads();

    // ---- Phase E: block-level strict cross term + final output (wave 0) ----
    // cross = sum_{I<J} R_I * S_J = sum_J S_J * prefixR_J  (512-element scan)
    if (wave == 0) {
        const int cb = lane << 4;   // 16 blocks per lane
        float locR = 0.f;
#pragma unroll
        for (int t = 0; t < 16; ++t) locR += Rb[cb + t];
        float incl = locR;
#pragma unroll
        for (int off = 1; off < 32; off <<= 1) {
            const float y = __shfl_up(incl, off, 32);
            if (lane >= off) incl += y;
        }
        float run    = incl - locR;   // exclusive prefix of R over previous lanes' chunks
        float crossp = 0.f;
#pragma unroll
        for (int t = 0; t < 16; ++t) {
            const int J = cb + t;
            crossp += Sb[J] * run;
            run    += Rb[J];
        }
#pragma unroll
        for (int off = 16; off >= 1; off >>= 1)
            crossp += __shfl_xor(crossp, off, 32);
        if (lane == 0) {
            float diag = 0.f;
#pragma unroll
            for (int w = 0; w < 32; ++w) diag += wpart[w];
            out[0] = diag + crossp;
        }
    }
}

extern "C" void kernel_launch(void* const* d_in, const int* in_sizes, int n_in,
                              void* d_out, int out_size, void* d_ws, size_t ws_size,
                              hipStream_t stream) {
    const float* scores = (const float*)d_in[0];   // [10M] f32
    const int*   target = (const int*)d_in[1];     // [10M] i32
    const int*   docs   = (const int*)d_in[2];     // [8192] i32
    (void)in_sizes; (void)n_in; (void)out_size; (void)d_ws; (void)ws_size;
    ranknet_fused_kernel<<<1, 1024, 0, stream>>>(scores, target, docs, (float*)d_out);
}